// TransformerScorer_41961830482227
// MI455X (gfx1250) — compile-verified
//
#include <hip/hip_runtime.h>
#include <hip/hip_bf16.h>
#include <math.h>

// Problem constants (from the reference)
#define B_ 8
#define N_ 4096
#define D_ 1024      // IN_FEATURES
#define H_ 1792      // HIDDEN_DIM

typedef __attribute__((ext_vector_type(2))) float v2f;
typedef __attribute__((ext_vector_type(8))) float v8f;

// ---------------------------------------------------------------------------
// 0) zero the scratch regions we accumulate into (ws is poisoned 0xAA)
// ---------------------------------------------------------------------------
__global__ void zero_ws_kernel(float* __restrict__ ws, int n) {
  int i = blockIdx.x * blockDim.x + threadIdx.x;
  if (i < n) ws[i] = 0.0f;
}

// ---------------------------------------------------------------------------
// 1) xs[b,d] = sum_n x[b,n,d]   (one pass over x: 128 MB)
//    grid (B, 8 n-chunks) x 256 threads; each thread owns 4 consecutive d
//    columns (float4 / global_load_b128, 512B per wave-instruction) over
//    512 rows, then atomicAdd's its partials (GLOBAL_ATOMIC_ADD_F32).
// ---------------------------------------------------------------------------
__global__ void colsum_x_kernel(const float* __restrict__ x, float* __restrict__ xs) {
  const int b = blockIdx.x;
  const int chunk = blockIdx.y;          // 512 rows per chunk
  const int d4 = threadIdx.x;            // 0..255 -> d = d4*4 .. d4*4+3
  const float4* xp = (const float4*)(x + ((size_t)b * N_ + (size_t)chunk * 512) * D_) + d4;
  float4 s = {0.f, 0.f, 0.f, 0.f};
  #pragma unroll 4
  for (int n = 0; n < 512; ++n) {
    float4 t = xp[(size_t)n * (D_ / 4)];
    s.x += t.x; s.y += t.y; s.z += t.z; s.w += t.w;
  }
  float* dst = &xs[b * D_ + d4 * 4];
  atomicAdd(dst + 0, s.x);
  atomicAdd(dst + 1, s.y);
  atomicAdd(dst + 2, s.z);
  atomicAdd(dst + 3, s.w);
}

// ---------------------------------------------------------------------------
// 2) ksum[b,h] = xs[b,:] . Wk[h,:] + N*bk[h]
//    WMMA f32 16x16x4: A = XS16 (rows 0..7 = xs[b], 8..15 = 0), B[k][n] = Wk[ht*16+n][k0+k].
//    A layout: lane L holds (M=L&15, K=(L>>4)*2 + vgpr) ; B symmetric.
//    D layout: VGPR i -> lanes 0..15 = (M=i, N=lane). Only M<8 is real.
//    grid (112 h-tiles, 4 k-splits) x 1 wave; partials atomicAdd'ed.
// ---------------------------------------------------------------------------
__global__ void gemm_ksum_kernel(const float* __restrict__ xs,
                                 const float* __restrict__ Wk,
                                 const float* __restrict__ bk,
                                 float* __restrict__ ksum) {
  const int ht   = blockIdx.x;           // h tile: columns ht*16..ht*16+15
  const int ks   = blockIdx.y;           // k split: [ks*256, ks*256+256)
  const int lane = threadIdx.x;          // 0..31 (wave32)
  const int m    = lane & 15;
  const int kb   = (lane >> 4) * 2;      // lanes 16..31 hold K=2,3
  const float amask = (m < 8) ? 1.0f : 0.0f;     // pad M=8..15 with zeros (no EXEC divergence)
  const float* arow = xs + (size_t)(m & 7) * D_;
  const float* brow = Wk + (size_t)(ht * 16 + m) * D_;   // lane's B column n == m

  v8f acc = {};
  const int kend = ks * 256 + 256;
  for (int k0 = ks * 256; k0 < kend; k0 += 4) {
    v2f a, b;
    a.x = arow[k0 + kb]     * amask;
    a.y = arow[k0 + kb + 1] * amask;
    b.x = brow[k0 + kb];
    b.y = brow[k0 + kb + 1];
    acc = __builtin_amdgcn_wmma_f32_16x16x4_f32(
        /*neg_a=*/false, a, /*neg_b=*/false, b,
        /*c_mod=*/(short)0, acc, /*reuse_a=*/false, /*reuse_b=*/false);
  }

  if (lane < 16) {
    const int h = ht * 16 + lane;
    const float bias = (ks == 0) ? (float)N_ * bk[h] : 0.0f;   // same bias for every b
    #pragma unroll
    for (int i = 0; i < 8; ++i)                 // VGPR i, lanes 0..15 -> M=i (valid rows)
      atomicAdd(&ksum[i * H_ + h], acc[i] + bias);
  }
}

// ---------------------------------------------------------------------------
// 3) c[b] = bq . ksum[b,:]   (tiny; bq is zeros at init but handle generally)
// ---------------------------------------------------------------------------
__global__ void dot_c_kernel(const float* __restrict__ ksum,
                             const float* __restrict__ bq,
                             float* __restrict__ c) {
  __shared__ float red[256];
  const int b = blockIdx.x;
  float s = 0.0f;
  for (int h = threadIdx.x; h < H_; h += 256) s += bq[h] * ksum[b * H_ + h];
  red[threadIdx.x] = s;
  __syncthreads();
  for (int off = 128; off > 0; off >>= 1) {
    if (threadIdx.x < off) red[threadIdx.x] += red[threadIdx.x + off];
    __syncthreads();
  }
  if (threadIdx.x == 0) c[b] = red[0];
}

// ---------------------------------------------------------------------------
// 4) v[b,d] = sum_h ksum[b,h] * Wq[h,d]
//    WMMA f32 16x16x4: A = KSUM16 (rows 0..7 = ksum[b]), B[k][n] = Wq[k][dt*16+n]
//    (row-major Wq -> B loads are 64B-contiguous across lanes).
//    grid (64 d-tiles, 8 k-splits of 224) x 1 wave; partials atomicAdd'ed.
// ---------------------------------------------------------------------------
__global__ void gemm_v_kernel(const float* __restrict__ ksum,
                              const float* __restrict__ Wq,
                              float* __restrict__ v) {
  const int dt   = blockIdx.x;
  const int ks   = blockIdx.y;           // k split: [ks*224, ks*224+224)
  const int lane = threadIdx.x;
  const int m    = lane & 15;
  const int kb   = (lane >> 4) * 2;
  const float amask = (m < 8) ? 1.0f : 0.0f;
  const float* arow = ksum + (size_t)(m & 7) * H_;
  const int ncol = dt * 16 + m;          // lane's B column

  v8f acc = {};
  const int kend = ks * 224 + 224;
  for (int k0 = ks * 224; k0 < kend; k0 += 4) {
    v2f a, b;
    a.x = arow[k0 + kb]     * amask;
    a.y = arow[k0 + kb + 1] * amask;
    b.x = Wq[(size_t)(k0 + kb)     * D_ + ncol];
    b.y = Wq[(size_t)(k0 + kb + 1) * D_ + ncol];
    acc = __builtin_amdgcn_wmma_f32_16x16x4_f32(
        false, a, false, b, (short)0, acc, false, false);
  }

  if (lane < 16) {
    const int d = dt * 16 + lane;
    #pragma unroll
    for (int i = 0; i < 8; ++i)
      atomicAdd(&v[i * D_ + d], acc[i]);
  }
}

// ---------------------------------------------------------------------------
// 5) scores[b,n] = (x[b,n,:] . v[b,:] + c[b]) / (N * sqrt(H))
//    Dominant pass (128 MB of x; hot in the 192 MB L2 after pass 1).
//    1 row per wave, 8 waves/block; v[b] staged in LDS; float4 per-lane loads
//    (global_load_b128, 512B per wave-instruction); wave32 __shfl_xor reduce.
// ---------------------------------------------------------------------------
__global__ void scores_kernel(const float* __restrict__ x,
                              const float* __restrict__ v,
                              const float* __restrict__ c,
                              float* __restrict__ out) {
  __shared__ float vs[D_];
  const int rows0 = blockIdx.x * 8;      // 8 rows per block, all in the same batch
  const int b = rows0 / N_;              // N_ % 8 == 0
  {
    const float4* vsrc = (const float4*)(v + b * D_);
    float4* vdst = (float4*)vs;
    for (int i = threadIdx.x; i < D_ / 4; i += 256) vdst[i] = vsrc[i];
  }
  __syncthreads();

  const int wave = threadIdx.x >> 5;
  const int lane = threadIdx.x & 31;
  const int row  = rows0 + wave;
  const float4* xp4 = (const float4*)(x + (size_t)row * D_);
  const float4* vs4 = (const float4*)vs;

  float s = 0.0f;
  #pragma unroll
  for (int i = lane; i < D_ / 4; i += 32) {   // 8 iterations, 8 b128 loads in flight
    float4 xv = xp4[i];
    float4 vv = vs4[i];
    s += xv.x * vv.x + xv.y * vv.y + xv.z * vv.z + xv.w * vv.w;
  }

  #pragma unroll
  for (int off = 16; off > 0; off >>= 1) s += __shfl_xor(s, off, 32);

  if (lane == 0) {
    const float scale = 1.0f / ((float)N_ * sqrtf((float)H_));
    out[row] = (s + c[b]) * scale;
  }
}

// ---------------------------------------------------------------------------
// launcher — inputs in setup_inputs() order: x, Wq, bq, Wk, bk (all f32)
// ---------------------------------------------------------------------------
extern "C" void kernel_launch(void* const* d_in, const int* in_sizes, int n_in,
                              void* d_out, int out_size, void* d_ws, size_t ws_size,
                              hipStream_t stream) {
  const float* x  = (const float*)d_in[0];
  const float* Wq = (const float*)d_in[1];
  const float* bq = (const float*)d_in[2];
  const float* Wk = (const float*)d_in[3];
  const float* bk = (const float*)d_in[4];
  float* out = (float*)d_out;

  // scratch layout (floats): xs[8192] | ksum[14336] | v[8192] | c[8]
  float* ws   = (float*)d_ws;
  float* xs   = ws;
  float* ksum = ws + (B_ * D_);
  float* v    = ws + (B_ * D_) + (B_ * H_);
  float* c    = ws + (B_ * D_) + (B_ * H_) + (B_ * D_);
  const int ztot = (B_ * D_) + (B_ * H_) + (B_ * D_) + B_;

  zero_ws_kernel<<<(ztot + 255) / 256, 256, 0, stream>>>(ws, ztot);
  colsum_x_kernel<<<dim3(B_, 8), 256, 0, stream>>>(x, xs);
  gemm_ksum_kernel<<<dim3(H_ / 16, 4), 32, 0, stream>>>(xs, Wk, bk, ksum);
  dot_c_kernel<<<B_, 256, 0, stream>>>(ksum, bq, c);
  gemm_v_kernel<<<dim3(D_ / 16, 8), 32, 0, stream>>>(ksum, Wq, v);
  scores_kernel<<<(B_ * N_) / 8, 256, 0, stream>>>(x, v, c, out);
}